// clDice_41729902248393
// MI455X (gfx1250) — compile-verified
//
#include <hip/hip_runtime.h>

#define IMG   1024
#define TSZ   32
#define HALO  5
#define S0    (TSZ + 2*HALO)   // 42
#define STRD  44               // LDS row stride (floats)
#define NTHR  256

// ---------------- CDNA5 async global->LDS path (probe-verified live) ----------------
#if defined(__has_builtin)
#  if __has_builtin(__builtin_amdgcn_global_load_async_to_lds_b32)
#    define HAVE_ASYNC_LDS 1
#  endif
#endif
#ifndef HAVE_ASYNC_LDS
#  define HAVE_ASYNC_LDS 0
#endif

typedef __attribute__((address_space(1))) int gas_i32;
typedef __attribute__((address_space(3))) int las_i32;

__device__ __forceinline__ void cp_g2l_b32(float* l, const float* g) {
#if HAVE_ASYNC_LDS
  __builtin_amdgcn_global_load_async_to_lds_b32((gas_i32*)g, (las_i32*)l, 0, 0);
#else
  *l = *g;
#endif
}

__device__ __forceinline__ void wait_async_lds() {
#if HAVE_ASYNC_LDS
#  if defined(__has_builtin) && __has_builtin(__builtin_amdgcn_s_wait_asynccnt)
  __builtin_amdgcn_s_wait_asynccnt(0);
#  else
  asm volatile("s_wait_asynccnt 0" ::: "memory");
#  endif
#endif
}

// ---------------- math helpers ----------------
__device__ __forceinline__ float gelu_f(float x) {
  // exact (erf-based) GELU, matches jax.nn.gelu(approximate=False)
  return 0.5f * x * (1.0f + erff(x * 0.70710678118654752440f));
}

// ---------------- tile staging ----------------
template <bool BORDER>
__device__ __forceinline__ void stage_tile(const float* __restrict__ img,
                                           float* __restrict__ A,
                                           int gx0, int gy0, int tid) {
  for (int i = tid; i < S0 * S0; i += NTHR) {
    const int by = i / S0, bx = i % S0;
    const int gy = gy0 + by - HALO, gx = gx0 + bx - HALO;
    float* l = &A[by * STRD + bx];
    if (BORDER) {
      if (((unsigned)gy < (unsigned)IMG) & ((unsigned)gx < (unsigned)IMG))
        cp_g2l_b32(l, img + (size_t)gy * IMG + gx);
      else
        *l = __builtin_inff();  // +inf = erode (min) identity outside image
    } else {
      cp_g2l_b32(l, img + (size_t)gy * IMG + gx);
    }
  }
}

// plus-shaped (5-point) min erode over region [K, S0-K)^2; +inf outside image
template <bool BORDER, int K>
__device__ __forceinline__ void erode_pass(const float* __restrict__ src,
                                           float* __restrict__ dst,
                                           int gx0, int gy0, int tid) {
  constexpr int S = S0 - 2 * K;
  constexpr int N = S * S;
  for (int i = tid; i < N; i += NTHR) {
    const int by = K + i / S;
    const int bx = K + i % S;
    const int o  = by * STRD + bx;
    float v = fminf(fminf(src[o - STRD], src[o + STRD]),
                    fminf(src[o - 1],    src[o + 1]));
    v = fminf(src[o], v);
    if (BORDER) {
      const int gy = gy0 + by - HALO, gx = gx0 + bx - HALO;
      const bool in = ((unsigned)gy < (unsigned)IMG) & ((unsigned)gx < (unsigned)IMG);
      dst[o] = in ? v : __builtin_inff();
    } else {
      dst[o] = v;
    }
  }
}

// 3x3 max dilate at one pixel; OOB neighbors excluded (treated as -inf)
template <bool BORDER>
__device__ __forceinline__ float dil3x3(const float* __restrict__ buf,
                                        int cy, int cx, int gy, int gx) {
  float m = -__builtin_inff();
#pragma unroll
  for (int dy = -1; dy <= 1; ++dy) {
#pragma unroll
    for (int dx = -1; dx <= 1; ++dx) {
      const float v = buf[(cy + dy) * STRD + (cx + dx)];
      if (BORDER) {
        const bool in = ((unsigned)(gy + dy) < (unsigned)IMG) &
                        ((unsigned)(gx + dx) < (unsigned)IMG);
        m = fmaxf(m, in ? v : -__builtin_inff());
      } else {
        m = fmaxf(m, v);
      }
    }
  }
  return m;
}

// ---------------- full soft_skel(img, 3) for one tile ----------------
template <bool BORDER>
__device__ __forceinline__ void skel_pass(const float* __restrict__ img,
                                          float* __restrict__ A,
                                          float* __restrict__ B,
                                          int gx0, int gy0, int tid,
                                          int lx, int ly0,
                                          const float c0[4], float skel[4]) {
  __syncthreads();  // all prior readers of A/B finished
  stage_tile<BORDER>(img, A, gx0, gy0, tid);
  wait_async_lds();
  __syncthreads();

  float eprev[4];
#pragma unroll
  for (int j = 0; j < 4; ++j) eprev[j] = c0[j];

  // e1 = erode(img); skel = gelu(img - dilate(e1))
  erode_pass<BORDER, 1>(A, B, gx0, gy0, tid);
  __syncthreads();
#pragma unroll
  for (int j = 0; j < 4; ++j) {
    const int cy = ly0 + 8 * j + HALO, cx = lx + HALO;
    const int gy = gy0 + ly0 + 8 * j, gx = gx0 + lx;
    const float d = dil3x3<BORDER>(B, cy, cx, gy, gx);
    skel[j]  = gelu_f(eprev[j] - d);
    eprev[j] = B[cy * STRD + cx];
  }
  // iter 1: e2; delta = gelu(e1 - dilate(e2)); skel += gelu(delta - skel*delta)
  erode_pass<BORDER, 2>(B, A, gx0, gy0, tid);
  __syncthreads();
#pragma unroll
  for (int j = 0; j < 4; ++j) {
    const int cy = ly0 + 8 * j + HALO, cx = lx + HALO;
    const int gy = gy0 + ly0 + 8 * j, gx = gx0 + lx;
    const float d  = dil3x3<BORDER>(A, cy, cx, gy, gx);
    const float dl = gelu_f(eprev[j] - d);
    const float s  = skel[j];
    skel[j]  = s + gelu_f(dl - s * dl);
    eprev[j] = A[cy * STRD + cx];
  }
  // iter 2: e3
  erode_pass<BORDER, 3>(A, B, gx0, gy0, tid);
  __syncthreads();
#pragma unroll
  for (int j = 0; j < 4; ++j) {
    const int cy = ly0 + 8 * j + HALO, cx = lx + HALO;
    const int gy = gy0 + ly0 + 8 * j, gx = gx0 + lx;
    const float d  = dil3x3<BORDER>(B, cy, cx, gy, gx);
    const float dl = gelu_f(eprev[j] - d);
    const float s  = skel[j];
    skel[j]  = s + gelu_f(dl - s * dl);
    eprev[j] = B[cy * STRD + cx];
  }
  // iter 3: e4
  erode_pass<BORDER, 4>(B, A, gx0, gy0, tid);
  __syncthreads();
#pragma unroll
  for (int j = 0; j < 4; ++j) {
    const int cy = ly0 + 8 * j + HALO, cx = lx + HALO;
    const int gy = gy0 + ly0 + 8 * j, gx = gx0 + lx;
    const float d  = dil3x3<BORDER>(A, cy, cx, gy, gx);
    const float dl = gelu_f(eprev[j] - d);
    const float s  = skel[j];
    skel[j] = s + gelu_f(dl - s * dl);
  }
}

// ---------------- kernels ----------------
__global__ void cldice_zero_acc(float* acc) {
  const int i = threadIdx.x;
  if (i < 52) acc[i] = 0.0f;
}

__global__ __launch_bounds__(NTHR) void cldice_main(
    const float* __restrict__ pred, const float* __restrict__ tru,
    float* __restrict__ acc) {
  __shared__ float A[S0 * STRD];
  __shared__ float B[S0 * STRD];
  __shared__ float red[7 * 8];

  const int gx0 = blockIdx.x * TSZ;
  const int gy0 = blockIdx.y * TSZ;
  const int nb  = blockIdx.z;
  const size_t base = (size_t)nb * (size_t)(IMG * IMG);
  const int tid = threadIdx.x;
  const int lx  = tid & (TSZ - 1);
  const int ly0 = tid >> 5;  // 0..7; each thread owns rows ly0 + 8*j

  float predc[4], truc[4];
#pragma unroll
  for (int j = 0; j < 4; ++j) {
    const int gy = gy0 + ly0 + 8 * j, gx = gx0 + lx;
    const size_t off = base + (size_t)gy * IMG + gx;
    predc[j] = pred[off];
    truc[j]  = tru[off];
  }

  // tiles whose 5-wide halo can leave the image: only outermost ring (12% of tiles)
  const bool border = (blockIdx.x == 0) || (blockIdx.x == gridDim.x - 1) ||
                      (blockIdx.y == 0) || (blockIdx.y == gridDim.y - 1);

  float skelp[4], skelt[4];
  if (border) {
    skel_pass<true>(pred + base, A, B, gx0, gy0, tid, lx, ly0, predc, skelp);
    skel_pass<true>(tru  + base, A, B, gx0, gy0, tid, lx, ly0, truc,  skelt);
  } else {
    skel_pass<false>(pred + base, A, B, gx0, gy0, tid, lx, ly0, predc, skelp);
    skel_pass<false>(tru  + base, A, B, gx0, gy0, tid, lx, ly0, truc,  skelt);
  }

  // s0=SUM skelp*t  s1=SUM skelp  s2=SUM skelt*p  s3=SUM skelt
  // s4=SUM sig(p)*t s5=SUM sig(p) s6=SUM t        (s4..s6 per-batch)
  float s0 = 0, s1 = 0, s2 = 0, s3 = 0, s4 = 0, s5 = 0, s6 = 0;
#pragma unroll
  for (int j = 0; j < 4; ++j) {
    s1 += skelp[j];
    s0 += skelp[j] * truc[j];
    s3 += skelt[j];
    s2 += skelt[j] * predc[j];
    const float sg = 1.0f / (1.0f + expf(-predc[j]));
    s5 += sg;
    s4 += sg * truc[j];
    s6 += truc[j];
  }

  // workgroup reduction: wave32 shuffle, then per-wave partials in LDS
  const int lane = tid & 31, wave = tid >> 5;
  float vals[7] = {s0, s1, s2, s3, s4, s5, s6};
#pragma unroll
  for (int i = 0; i < 7; ++i) {
    float v = vals[i];
#pragma unroll
    for (int o = 16; o > 0; o >>= 1) v += __shfl_down(v, o, 32);
    if (lane == 0) red[i * 8 + wave] = v;
  }
  __syncthreads();
  if (tid == 0) {
    float t[7];
#pragma unroll
    for (int i = 0; i < 7; ++i) {
      float v = 0.0f;
#pragma unroll
      for (int w = 0; w < 8; ++w) v += red[i * 8 + w];
      t[i] = v;
    }
    atomicAdd(&acc[0], t[0]);
    atomicAdd(&acc[1], t[1]);
    atomicAdd(&acc[2], t[2]);
    atomicAdd(&acc[3], t[3]);
    atomicAdd(&acc[4 + 3 * nb + 0], t[4]);  // inter_n
    atomicAdd(&acc[4 + 3 * nb + 1], t[5]);  // sum sigmoid(pred)_n
    atomicAdd(&acc[4 + 3 * nb + 2], t[6]);  // sum true_n
  }
}

__global__ void cldice_finalize(const float* __restrict__ acc,
                                float* __restrict__ out) {
  const float tprec = (acc[0] + 1.0f) / (acc[1] + 1.0f);
  const float tsens = (acc[2] + 1.0f) / (acc[3] + 1.0f);
  const float cl = 1.0f - 2.0f * (tprec * tsens) / (tprec + tsens);
  float dsum = 0.0f;
  for (int n = 0; n < 16; ++n) {
    const float inter = acc[4 + 3 * n + 0];
    const float sp    = acc[4 + 3 * n + 1];
    const float st    = acc[4 + 3 * n + 2];
    const float gd = (2.0f * inter + 1e-4f) / (sp + st + 1e-4f);
    dsum += 1.0f - gd;
  }
  const float dice = dsum * (1.0f / 16.0f);
  out[0] = 0.5f * cl + 0.5f * dice;
}

extern "C" void kernel_launch(void* const* d_in, const int* in_sizes, int n_in,
                              void* d_out, int out_size, void* d_ws, size_t ws_size,
                              hipStream_t stream) {
  const float* pred = (const float*)d_in[0];
  const float* tru  = (const float*)d_in[1];
  float* acc = (float*)d_ws;   // 52 floats
  float* out = (float*)d_out;  // 1 float

  cldice_zero_acc<<<1, 64, 0, stream>>>(acc);
  dim3 grid(IMG / TSZ, IMG / TSZ, 16);
  cldice_main<<<grid, NTHR, 0, stream>>>(pred, tru, acc);
  cldice_finalize<<<1, 1, 0, stream>>>(acc, out);
}